// Clustering_loss_52037823758794
// MI455X (gfx1250) — compile-verified
//
#include <hip/hip_runtime.h>
#include <math.h>

typedef float v2f __attribute__((ext_vector_type(2)));
typedef float v8f __attribute__((ext_vector_type(8)));

#define PADV 50
#define HH 512
#define WW 512
#define CC 19
#define KP 20          // channels padded to multiple of 4 for K=4 WMMA steps
#define SS 8192
#define NN 2
#define CROPW 412      // 512 - 2*50
#define NBLK (SS / 16) // 512 blocks of 16 rows/cols

// ---------------------------------------------------------------------------
// Stage 1: gather sampled pixel vectors into Pg[n][s][c], c padded to KP=20
// ---------------------------------------------------------------------------
__global__ __launch_bounds__(256) void gather_kernel(const float* __restrict__ out,
                                                     const int* __restrict__ sample,
                                                     float* __restrict__ Pg) {
    int t = blockIdx.x * 256 + threadIdx.x;
    const int total = NN * SS * KP;
    if (t >= total) return;
    int cp = t % KP;
    int ns = t / KP;
    int s  = ns % SS;
    int n  = ns / SS;
    float v = 0.0f;
    if (cp < CC) {
        int f   = sample[s];
        int r   = f / CROPW;
        int col = f - r * CROPW;
        size_t off = (((size_t)(n * CC + cp)) * HH + (PADV + r)) * WW + (PADV + col);
        v = out[off];
    }
    Pg[t] = v;
}

// ---------------------------------------------------------------------------
// Stage 2: sim = P * P^T via V_WMMA_F32_16X16X4_F32, fused row-argmax.
// One wave handles one (n, 16-row block), sweeping all 512 column blocks with
// a double-buffered (software-pipelined) B-fragment stream.
// ---------------------------------------------------------------------------
__device__ __forceinline__ void load_frags(const float* __restrict__ brow, v2f b[5]) {
#pragma unroll
    for (int t = 0; t < 5; ++t) {
        b[t].x = brow[4 * t + 0];
        b[t].y = brow[4 * t + 1];
    }
}

__device__ __forceinline__ void tile_update(const v2f a[5], const v2f b[5], int col,
                                            float best[8], int bidx[8]) {
    v8f c = {0.f, 0.f, 0.f, 0.f, 0.f, 0.f, 0.f, 0.f};
#pragma unroll
    for (int t = 0; t < 5; ++t)
        c = __builtin_amdgcn_wmma_f32_16x16x4_f32(
                false, a[t], false, b[t], (short)0, c, false, false);
#pragma unroll
    for (int r = 0; r < 8; ++r) {   // VGPR r: rows r / r+8 in the two lane halves
        float v = c[r];
        if (v > best[r]) { best[r] = v; bidx[r] = col; }  // strict > keeps earliest j
    }
}

__global__ __launch_bounds__(256) void simargmax_kernel(const float* __restrict__ Pg,
                                                        int* __restrict__ ind) {
    const int lane = threadIdx.x & 31;
    const int wave = blockIdx.x * 8 + (threadIdx.x >> 5);
    const int n    = wave / NBLK;
    const int i    = wave - n * NBLK;
    const int half = lane >> 4;   // 0: K pair {k,k+1}; 1: {k+2,k+3}
    const int l15  = lane & 15;

    const float* base = Pg + (size_t)n * SS * KP;

    // A fragments for row block i (loop invariant)
    const float* arow = base + (size_t)(i * 16 + l15) * KP + 2 * half;
    v2f a[5];
    load_frags(arow, a);

    float best[8];
    int   bidx[8];
#pragma unroll
    for (int r = 0; r < 8; ++r) { best[r] = -INFINITY; bidx[r] = 0; }

    // Double-buffered sweep over column blocks: loads for block j+1 are issued
    // before the WMMAs consuming block j, hiding L1/L2 latency at 1 wave/SIMD.
    const float* brow0 = base + (size_t)l15 * KP + 2 * half;
    v2f b0[5], b1[5];
    load_frags(brow0, b0);
    for (int j = 0; j < NBLK; j += 2) {
        load_frags(brow0 + (size_t)(j + 1) * 16 * KP, b1);
        tile_update(a, b0, j * 16 + l15, best, bidx);
        // For j+2 == NBLK this reads the padding block after Pg (never consumed).
        load_frags(brow0 + (size_t)(j + 2) * 16 * KP, b0);
        tile_update(a, b1, (j + 1) * 16 + l15, best, bidx);
    }

    // Reduce across the 16 lanes of each half (xor masks < 16 stay in-half).
    // Tie-break: smaller column index wins (jnp.argmax takes first occurrence).
#pragma unroll
    for (int r = 0; r < 8; ++r) {
#pragma unroll
        for (int off = 1; off < 16; off <<= 1) {
            float ov = __shfl_xor(best[r], off, 32);
            int   oi = __shfl_xor(bidx[r], off, 32);
            if (ov > best[r] || (ov == best[r] && oi < bidx[r])) {
                best[r] = ov; bidx[r] = oi;
            }
        }
    }

    if (l15 == 0) {
        const int rowbase = i * 16 + 8 * half;
#pragma unroll
        for (int r = 0; r < 8; ++r)
            ind[n * SS + rowbase + r] = bidx[r];
    }
}

// ---------------------------------------------------------------------------
// Stage 3: loss partials. loss[n][s] = sum_c -(P[ind][c] * log(P[s][c]+eps))
// Deterministic LDS tree reduction per block -> 64 partials.
// ---------------------------------------------------------------------------
__global__ __launch_bounds__(256) void loss_kernel(const float* __restrict__ Pg,
                                                   const int* __restrict__ ind,
                                                   float* __restrict__ partials) {
    __shared__ float sm[256];
    int t = blockIdx.x * 256 + threadIdx.x;  // t in [0, NN*SS)
    int n = t / SS;
    int s = t - n * SS;
    const float* prow = Pg + ((size_t)n * SS + s) * KP;
    const float* qrow = Pg + ((size_t)n * SS + ind[t]) * KP;
    float acc = 0.0f;
#pragma unroll
    for (int c = 0; c < CC; ++c)
        acc -= qrow[c] * logf(prow[c] + 1e-6f);
    sm[threadIdx.x] = acc;
    __syncthreads();
    for (int stp = 128; stp > 0; stp >>= 1) {
        if (threadIdx.x < stp) sm[threadIdx.x] += sm[threadIdx.x + stp];
        __syncthreads();
    }
    if (threadIdx.x == 0) partials[blockIdx.x] = sm[0];
}

// ---------------------------------------------------------------------------
// Stage 4: final fixed-order reduction + mean scale
// ---------------------------------------------------------------------------
__global__ __launch_bounds__(64) void finalize_kernel(const float* __restrict__ partials,
                                                      float* __restrict__ out) {
    __shared__ float sm[64];
    sm[threadIdx.x] = partials[threadIdx.x];
    __syncthreads();
    for (int stp = 32; stp > 0; stp >>= 1) {
        if (threadIdx.x < stp) sm[threadIdx.x] += sm[threadIdx.x + stp];
        __syncthreads();
    }
    if (threadIdx.x == 0)
        out[0] = sm[0] * (1.0f / (float)(NN * SS * CC));
}

// ---------------------------------------------------------------------------
extern "C" void kernel_launch(void* const* d_in, const int* in_sizes, int n_in,
                              void* d_out, int out_size, void* d_ws, size_t ws_size,
                              hipStream_t stream) {
    const float* output = (const float*)d_in[0];   // (2,19,512,512) f32
    const int*   sample = (const int*)d_in[1];     // (8192,) int32 (jax x64 off)
    float*       outp   = (float*)d_out;           // (1,) f32

    // Workspace layout (with one 16-row pad block after Pg for the pipelined
    // prologue over-read; its contents are never consumed).
    float* Pg       = (float*)d_ws;                                  // NN*SS*KP floats
    float* PgEnd    = Pg + (size_t)NN * SS * KP + 16 * KP;           // +pad block
    int*   ind      = (int*)PgEnd;                                   // NN*SS ints
    float* partials = (float*)(ind + (size_t)NN * SS);               // 64 floats

    // Stage 1: gather (327,680 threads)
    {
        int total  = NN * SS * KP;
        int blocks = (total + 255) / 256;
        gather_kernel<<<blocks, 256, 0, stream>>>(output, sample, Pg);
    }
    // Stage 2: WMMA Gram + argmax (1024 waves = 128 blocks x 8 waves)
    simargmax_kernel<<<(NN * NBLK) / 8, 256, 0, stream>>>(Pg, ind);
    // Stage 3: loss partials (16384 threads = 64 blocks)
    loss_kernel<<<(NN * SS) / 256, 256, 0, stream>>>(Pg, ind, partials);
    // Stage 4: finalize
    finalize_kernel<<<1, 64, 0, stream>>>(partials, outp);
}